// CEHawkesProcess_6442450944058
// MI455X (gfx1250) — compile-verified
//
#include <hip/hip_runtime.h>
#include <math.h>

#define NTY 50
#define NCA 20
#define EDIM 64
#define BATCH 4
#define SEQL 256
#define WPB 8            // waves per block
#define TPB (WPB * 32)

typedef __attribute__((ext_vector_type(2))) float v2f;
typedef __attribute__((ext_vector_type(8))) float v8f;

// ---------------------------------------------------------------------------
// Wave-wide reduction of 64 values (2 per lane: elements e0=2*lane, 2*lane+1)
// via V_WMMA_F32_16X16X4_F32 with B = ones:
//   A(16x4) layout: VGPR0 = {lanes0-15: K=0, lanes16-31: K=2},
//                   VGPR1 = {lanes0-15: K=1, lanes16-31: K=3}
// so a per-lane v2f holds exactly one row-chunk of A; D row-sums land in the
// accumulator (row m replicated across columns). Summing the 8 acc VGPRs in a
// lane gives rows 0-7 (lanes<16) / rows 8-15 (lanes>=16); one xor-16 shuffle
// finishes the reduction. EXEC must be all ones at the call site.
// ---------------------------------------------------------------------------
__device__ __forceinline__ float wave_sum64_wmma(float p0, float p1) {
  v2f a;
  a.x = p0;
  a.y = p1;
  v2f bones;
  bones.x = 1.0f;
  bones.y = 1.0f;
  v8f c = {};
  v8f d = __builtin_amdgcn_wmma_f32_16x16x4_f32(
      /*neg_a=*/false, a, /*neg_b=*/false, bones,
      /*c_mod=*/(short)0, c, /*reuse_a=*/false, /*reuse_b=*/false);
  float s = d[0] + d[1] + d[2] + d[3] + d[4] + d[5] + d[6] + d[7];
  s += __shfl_xor(s, 16, 32);
  return s;
}

struct Smem {
  float temb[NTY * EDIM];   // 12.8 KB
  float cemb[NCA * EDIM];   //  5.1 KB
  float times[SEQL];
  int   types[SEQL];
  int   cats[SEQL];
};

__device__ __forceinline__ void stage_smem(Smem& sm, int b,
                                           const float* __restrict__ g_times,
                                           const int* __restrict__ g_types,
                                           const int* __restrict__ g_cats,
                                           const float* __restrict__ temb,
                                           const float* __restrict__ cemb) {
  for (int idx = threadIdx.x; idx < NTY * EDIM; idx += TPB) sm.temb[idx] = temb[idx];
  for (int idx = threadIdx.x; idx < NCA * EDIM; idx += TPB) sm.cemb[idx] = cemb[idx];
  for (int idx = threadIdx.x; idx < SEQL; idx += TPB) {
    sm.times[idx] = g_times[b * SEQL + idx];
    sm.types[idx] = g_types[b * SEQL + idx];
    sm.cats[idx]  = g_cats[b * SEQL + idx];
  }
  __syncthreads();
}

// ---------------------------------------------------------------------------
// Kernel 1: one wave per event. Computes ll_terms[b*L+i] = log(lam)+lam.
// ---------------------------------------------------------------------------
__global__ __launch_bounds__(TPB) void hawkes_event_ll(
    const float* __restrict__ g_times, const int* __restrict__ g_types,
    const int* __restrict__ g_cats, const float* __restrict__ temb,
    const float* __restrict__ cemb, const float* __restrict__ ga,
    const float* __restrict__ gb, const float* __restrict__ gA,
    const float* __restrict__ gP, const float* __restrict__ gBm,
    const float* __restrict__ gQ, float* __restrict__ ll_terms) {
  __shared__ Smem sm;
  const int blocksPerSeq = SEQL / WPB;  // 32
  const int b  = blockIdx.x / blocksPerSeq;
  const int i0 = (blockIdx.x % blocksPerSeq) * WPB;
  stage_smem(sm, b, g_times, g_types, g_cats, temb, cemb);

  const int wave = threadIdx.x >> 5;
  const int lane = threadIdx.x & 31;
  const int i    = i0 + wave;
  const int t    = sm.types[i];
  const float ti = sm.times[i];
  const int e0   = lane * 2;

  const v2f f_t = *(const v2f*)&sm.temb[t * EDIM + e0];
  const v2f a_t = *(const v2f*)&ga[t * EDIM + e0];

  float lam;
  if (i == 0) {
    // empty-history branch: sum_e softplus(f*a)
    float p0 = logf(1.0f + __expf(f_t.x * a_t.x));
    float p1 = logf(1.0f + __expf(f_t.y * a_t.y));
    lam = wave_sum64_wmma(p0, p1);
  } else {
    const int cl   = sm.cats[i - 1];
    const v2f b_cl = *(const v2f*)&gb[cl * EDIM + e0];
    const v2f g_cl = *(const v2f*)&sm.cemb[cl * EDIM + e0];
    v2f macc; macc.x = 0.0f; macc.y = 0.0f;
    v2f nacc; nacc.x = 0.0f; nacc.y = 0.0f;
    for (int s = 0; s < i; ++s) {
      const float td = ti - sm.times[s];
      const int k  = sm.types[s];
      const int cs = sm.cats[s];
      const v2f f_k = *(const v2f*)&sm.temb[k * EDIM + e0];
      const v2f Akt = *(const v2f*)&gA[(k * NTY + t) * EDIM + e0];
      const v2f Pkt = *(const v2f*)&gP[(k * NTY + t) * EDIM + e0];
      const v2f g_s = *(const v2f*)&sm.cemb[cs * EDIM + e0];
      const v2f Bcc = *(const v2f*)&gBm[(cs * NCA + cl) * EDIM + e0];
      const v2f Qcc = *(const v2f*)&gQ[(cs * NCA + cl) * EDIM + e0];
      macc.x += f_t.x * Akt.x * f_k.x * __expf(-(f_t.x * Pkt.x * f_k.x) * td);
      macc.y += f_t.y * Akt.y * f_k.y * __expf(-(f_t.y * Pkt.y * f_k.y) * td);
      nacc.x += g_cl.x * Bcc.x * g_s.x * __expf(-(g_cl.x * Qcc.x * g_s.x) * td);
      nacc.y += g_cl.y * Bcc.y * g_s.y * __expf(-(g_cl.y * Qcc.y * g_s.y) * td);
    }
    const float p0 = f_t.x * (a_t.x + b_cl.x) + macc.x + f_t.x * nacc.x;
    const float p1 = f_t.y * (a_t.y + b_cl.y) + macc.y + f_t.y * nacc.y;
    lam = wave_sum64_wmma(p0, p1);
  }
  if (lane == 0) {
    const float c = (ti >= 0.0f) ? (logf(lam + 1e-16f) + lam) : 0.0f;
    ll_terms[b * SEQL + i] = c;
  }
}

// ---------------------------------------------------------------------------
// Kernel 2: per-sequence horizon intensity + integral term.
// integral_b = sum_t lam_t(T) * (T - times[L-1]) + I0 * times[0]
// ---------------------------------------------------------------------------
__global__ __launch_bounds__(TPB) void hawkes_integral(
    const float* __restrict__ g_times, const int* __restrict__ g_types,
    const int* __restrict__ g_cats, const int* __restrict__ gT,
    const float* __restrict__ temb, const float* __restrict__ cemb,
    const float* __restrict__ ga, const float* __restrict__ gb,
    const float* __restrict__ gA, const float* __restrict__ gP,
    const float* __restrict__ gBm, const float* __restrict__ gQ,
    float* __restrict__ seq_int) {
  __shared__ Smem sm;
  __shared__ float s_red[TPB];
  __shared__ float s_wsum[WPB];
  __shared__ float s_I0;
  const int b = blockIdx.x;
  stage_smem(sm, b, g_times, g_types, g_cats, temb, cemb);

  // I0 = sum softplus(type_emb * a) over NT*E — fixed-order tree (deterministic)
  float i0p = 0.0f;
  for (int idx = threadIdx.x; idx < NTY * EDIM; idx += TPB) {
    const float x = sm.temb[idx] * ga[idx];
    i0p += logf(1.0f + __expf(x));
  }
  s_red[threadIdx.x] = i0p;
  __syncthreads();
  for (int off = TPB / 2; off > 0; off >>= 1) {
    if (threadIdx.x < off) s_red[threadIdx.x] += s_red[threadIdx.x + off];
    __syncthreads();
  }
  if (threadIdx.x == 0) s_I0 = s_red[0];
  __syncthreads();

  const int wave = threadIdx.x >> 5;
  const int lane = threadIdx.x & 31;
  const int e0   = lane * 2;
  const int cl   = sm.cats[SEQL - 1];
  const float Tf = (float)(*gT);

  float wsum = 0.0f;
  for (int t = wave; t < NTY; t += WPB) {
    const v2f f_t  = *(const v2f*)&sm.temb[t * EDIM + e0];
    const v2f a_t  = *(const v2f*)&ga[t * EDIM + e0];
    const v2f b_cl = *(const v2f*)&gb[cl * EDIM + e0];
    v2f macc; macc.x = 0.0f; macc.y = 0.0f;
    for (int s = 0; s < SEQL; ++s) {
      const float td = Tf - sm.times[s];
      const int k = sm.types[s];
      const v2f f_k = *(const v2f*)&sm.temb[k * EDIM + e0];
      const v2f Akt = *(const v2f*)&gA[(k * NTY + t) * EDIM + e0];
      const v2f Pkt = *(const v2f*)&gP[(k * NTY + t) * EDIM + e0];
      macc.x += f_t.x * Akt.x * f_k.x * __expf(-(f_t.x * Pkt.x * f_k.x) * td);
      macc.y += f_t.y * Akt.y * f_k.y * __expf(-(f_t.y * Pkt.y * f_k.y) * td);
    }
    const float p0 = f_t.x * (a_t.x + b_cl.x) + macc.x;
    const float p1 = f_t.y * (a_t.y + b_cl.y) + macc.y;
    wsum += wave_sum64_wmma(p0, p1);
  }
  if (wave == 0) {
    // category channel folded: sum_t sum_e f[t,e]*n[cl,e] = sum_e colf[e]*n[cl,e]
    v2f colf; colf.x = 0.0f; colf.y = 0.0f;
    for (int t = 0; t < NTY; ++t) {
      const v2f ft = *(const v2f*)&sm.temb[t * EDIM + e0];
      colf.x += ft.x;
      colf.y += ft.y;
    }
    const v2f g_cl = *(const v2f*)&sm.cemb[cl * EDIM + e0];
    v2f nacc; nacc.x = 0.0f; nacc.y = 0.0f;
    for (int s = 0; s < SEQL; ++s) {
      const float td = Tf - sm.times[s];
      const int cs = sm.cats[s];
      const v2f g_s = *(const v2f*)&sm.cemb[cs * EDIM + e0];
      const v2f Bcc = *(const v2f*)&gBm[(cs * NCA + cl) * EDIM + e0];
      const v2f Qcc = *(const v2f*)&gQ[(cs * NCA + cl) * EDIM + e0];
      nacc.x += g_cl.x * Bcc.x * g_s.x * __expf(-(g_cl.x * Qcc.x * g_s.x) * td);
      nacc.y += g_cl.y * Bcc.y * g_s.y * __expf(-(g_cl.y * Qcc.y * g_s.y) * td);
    }
    wsum += wave_sum64_wmma(colf.x * nacc.x, colf.y * nacc.y);
  }
  if (lane == 0) s_wsum[wave] = wsum;
  __syncthreads();
  if (threadIdx.x == 0) {
    float sumLam = 0.0f;
    for (int w = 0; w < WPB; ++w) sumLam += s_wsum[w];
    seq_int[b] = sumLam * (Tf - sm.times[SEQL - 1]) + s_I0 * sm.times[0];
  }
}

// ---------------------------------------------------------------------------
// Kernel 3: deterministic final reduction. out = sum(integrals) - sum(lls)
// ---------------------------------------------------------------------------
__global__ __launch_bounds__(256) void hawkes_finalize(
    const float* __restrict__ ll_terms, const float* __restrict__ seq_int,
    float* __restrict__ out) {
  __shared__ float s_red[256];
  float p = 0.0f;
  for (int idx = threadIdx.x; idx < BATCH * SEQL; idx += 256) p += ll_terms[idx];
  s_red[threadIdx.x] = p;
  __syncthreads();
  for (int off = 128; off > 0; off >>= 1) {
    if (threadIdx.x < off) s_red[threadIdx.x] += s_red[threadIdx.x + off];
    __syncthreads();
  }
  if (threadIdx.x == 0) {
    float ints = 0.0f;
    for (int bb = 0; bb < BATCH; ++bb) ints += seq_int[bb];
    out[0] = ints - s_red[0];
  }
}

extern "C" void kernel_launch(void* const* d_in, const int* in_sizes, int n_in,
                              void* d_out, int out_size, void* d_ws, size_t ws_size,
                              hipStream_t stream) {
  const float* g_times = (const float*)d_in[0];
  const int*   g_types = (const int*)d_in[1];
  const int*   g_cats  = (const int*)d_in[2];
  const int*   gT      = (const int*)d_in[3];
  const float* temb    = (const float*)d_in[4];
  const float* cemb    = (const float*)d_in[5];
  const float* ga      = (const float*)d_in[6];
  const float* gb      = (const float*)d_in[7];
  const float* gA      = (const float*)d_in[8];
  const float* gP      = (const float*)d_in[9];
  const float* gBm     = (const float*)d_in[10];
  const float* gQ      = (const float*)d_in[11];

  float* ll_terms = (float*)d_ws;             // BATCH*SEQL floats
  float* seq_int  = ll_terms + BATCH * SEQL;  // BATCH floats

  hawkes_event_ll<<<BATCH * (SEQL / WPB), TPB, 0, stream>>>(
      g_times, g_types, g_cats, temb, cemb, ga, gb, gA, gP, gBm, gQ, ll_terms);
  hawkes_integral<<<BATCH, TPB, 0, stream>>>(
      g_times, g_types, g_cats, gT, temb, cemb, ga, gb, gA, gP, gBm, gQ, seq_int);
  hawkes_finalize<<<1, 256, 0, stream>>>(ll_terms, seq_int, (float*)d_out);
}